// Test_Backbone_33672543601351
// MI455X (gfx1250) — compile-verified
//
#include <hip/hip_runtime.h>

#define BN_EPS 1e-5f

typedef __attribute__((ext_vector_type(2))) float v2f;
typedef __attribute__((ext_vector_type(8))) float v8f;

// D = A(16x4 f32) * B(4x16 f32) + C(16x16 f32), full fp32 WMMA on CDNA5.
static __device__ __forceinline__ v8f mma4(v2f a, v2f b, v8f c) {
  return __builtin_amdgcn_wmma_f32_16x16x4_f32(false, a, false, b, (short)0, c,
                                               false, false);
}

// ---------------------------------------------------------------------------
// conv1: 7x7 stride-2 pad-3, 3->64 ch, fused BN + ReLU. Direct (K=147 is
// awkward for 16x16x4 tiling and this layer is <7% of total FLOPs).
// ---------------------------------------------------------------------------
__global__ __launch_bounds__(256) void k_conv7_bn_relu(
    const float* __restrict__ x, const float* __restrict__ w,
    const float* __restrict__ g, const float* __restrict__ b,
    const float* __restrict__ m, const float* __restrict__ v,
    float* __restrict__ out) {
  const int ox = blockIdx.x * 16 + threadIdx.x;
  const int oy = blockIdx.y * 16 + threadIdx.y;
  const int co = blockIdx.z;
  float acc = 0.f;
  const float* wp = w + co * 147;
  for (int ci = 0; ci < 3; ++ci) {
    const float* xp = x + ci * 512 * 512;
    const float* wc = wp + ci * 49;
    for (int ky = 0; ky < 7; ++ky) {
      int iy = oy * 2 + ky - 3;
      if (iy < 0 || iy >= 512) continue;
      for (int kx = 0; kx < 7; ++kx) {
        int ix = ox * 2 + kx - 3;
        if (ix < 0 || ix >= 512) continue;
        acc += xp[iy * 512 + ix] * wc[ky * 7 + kx];
      }
    }
  }
  float scale = g[co] / sqrtf(v[co] + BN_EPS);
  float y = acc * scale + (b[co] - m[co] * scale);
  out[co * 256 * 256 + oy * 256 + ox] = fmaxf(y, 0.f);
}

// ---------------------------------------------------------------------------
// maxpool 3x3 stride-2 pad-1 : 64x256x256 -> 64x128x128
// ---------------------------------------------------------------------------
__global__ __launch_bounds__(256) void k_maxpool(const float* __restrict__ in,
                                                 float* __restrict__ out) {
  int idx = blockIdx.x * blockDim.x + threadIdx.x;
  if (idx >= 64 * 128 * 128) return;
  int ox = idx & 127, oy = (idx >> 7) & 127, c = idx >> 14;
  const float* ip = in + c * 256 * 256;
  float best = -INFINITY;
  for (int ky = 0; ky < 3; ++ky) {
    int iy = oy * 2 + ky - 1;
    if (iy < 0 || iy >= 256) continue;
    for (int kx = 0; kx < 3; ++kx) {
      int ix = ox * 2 + kx - 1;
      if (ix < 0 || ix >= 256) continue;
      best = fmaxf(best, ip[iy * 256 + ix]);
    }
  }
  out[idx] = best;
}

// ---------------------------------------------------------------------------
// 3x3 conv as implicit GEMM on V_WMMA_F32_16X16X4_F32.
// Per wave: 16 output pixels (M) x 16 output channels (N).
// K is reordered as k = r*Ci + ci (r = 3x3 tap index): with r as the OUTER
// loop, each lane's spatial bounds test and (iy,ix) address are loop
// invariant, so the hot inner loop over ci is branch-free: out-of-bounds
// lanes read a zeroed scratch word via {base=zbuf, stride=0}.
// A frag: lanes 0-15 hold row M=lane, K = k..k+1 ; lanes 16-31 same rows,
// K = k+2..k+3 (ISA 32-bit A layout). Epilogue fuses BN / residual / ReLU
// and optionally copies the raw pre-BN conv output.
// ---------------------------------------------------------------------------
__global__ __launch_bounds__(256) void k_conv3_wmma(
    const float* __restrict__ in, const float* __restrict__ w,
    const float* __restrict__ g, const float* __restrict__ b,
    const float* __restrict__ m, const float* __restrict__ v,
    const float* __restrict__ idn, float* __restrict__ out,
    float* __restrict__ raw_out, const float* __restrict__ zbuf, int Ci, int H,
    int W, int Co, int Ho, int Wo, int stride, int relu) {
  const int lane = threadIdx.x & 31;
  const int wave = threadIdx.x >> 5;
  const int half = lane >> 4;
  const int l16 = lane & 15;
  const int HoWo = Ho * Wo;
  const int pixBase = (blockIdx.x * 8 + wave) * 16;  // wave-uniform
  const int coBase = blockIdx.y * 16;
  if (pixBase >= HoWo) return;  // whole wave exits together; EXEC stays full
  const long HW = (long)H * W;

  // this lane's A row (one output pixel)
  const int p = pixBase + l16;
  const bool pValid = (p < HoWo);
  const int oy = pValid ? (p / Wo) : 0;
  const int ox = pValid ? (p % Wo) : 0;
  const int iy0 = oy * stride - 1;
  const int ix0 = ox * stride - 1;

  const int kOff = half * 2;
  const long wrow = (long)(coBase + l16) * Ci * 9;  // B column (OIHW layout)

  v8f acc = {0.f, 0.f, 0.f, 0.f, 0.f, 0.f, 0.f, 0.f};
  for (int r = 0; r < 9; ++r) {
    const int ky = r / 3, kx = r - ky * 3;
    const int iy = iy0 + ky, ix = ix0 + kx;
    const bool ok = pValid && iy >= 0 && iy < H && ix >= 0 && ix < W;
    const long sA = ok ? HW : 0;                       // per-lane A stride
    const float* pa = (ok ? in + iy * W + ix : zbuf) + kOff * sA;
    const float* pb = w + wrow + kOff * 9 + r;         // stride 9 per K step
    for (int k0 = 0; k0 < Ci; k0 += 4) {
      v2f a, bb;
      a.x = pa[0];
      a.y = pa[sA];
      bb.x = pb[0];
      bb.y = pb[9];
      pa += 4 * sA;
      pb += 36;
      acc = mma4(a, bb, acc);
    }
  }

  // epilogue: C/D layout -> lane column = l16, rows i + 8*half
  const int ch = coBase + l16;
  float scale = g[ch] / sqrtf(v[ch] + BN_EPS);
  float shift = b[ch] - m[ch] * scale;
#pragma unroll
  for (int i = 0; i < 8; ++i) {
    int pp = pixBase + i + 8 * half;
    if (pp < HoWo) {
      float raw = acc[i];
      if (raw_out) raw_out[(long)ch * HoWo + pp] = raw;
      float y = raw * scale + shift;
      if (idn) y += idn[(long)ch * HoWo + pp];
      if (relu) y = fmaxf(y, 0.f);
      out[(long)ch * HoWo + pp] = y;
    }
  }
}

// ---------------------------------------------------------------------------
// 1x1 downsample conv (stride s) + BN, same WMMA tiling, K = Ci.
// Branch-free inner loop via the same zero-page trick for tail pixels.
// ---------------------------------------------------------------------------
__global__ __launch_bounds__(256) void k_conv1_wmma(
    const float* __restrict__ in, const float* __restrict__ w,
    const float* __restrict__ g, const float* __restrict__ b,
    const float* __restrict__ m, const float* __restrict__ v,
    float* __restrict__ out, const float* __restrict__ zbuf, int Ci, int H,
    int W, int Co, int Ho, int Wo, int stride) {
  const int lane = threadIdx.x & 31;
  const int wave = threadIdx.x >> 5;
  const int half = lane >> 4;
  const int l16 = lane & 15;
  const int HoWo = Ho * Wo;
  const int pixBase = (blockIdx.x * 8 + wave) * 16;
  const int coBase = blockIdx.y * 16;
  if (pixBase >= HoWo) return;
  const long HW = (long)H * W;

  const int p = pixBase + l16;
  const bool pValid = (p < HoWo);
  const int oy = pValid ? (p / Wo) : 0;
  const int ox = pValid ? (p % Wo) : 0;

  const int kOff = half * 2;
  const long sA = pValid ? HW : 0;
  const float* pa =
      (pValid ? in + (long)(oy * stride) * W + ox * stride : zbuf) + kOff * sA;
  const float* pb = w + (long)(coBase + l16) * Ci + kOff;

  v8f acc = {0.f, 0.f, 0.f, 0.f, 0.f, 0.f, 0.f, 0.f};
  for (int k0 = 0; k0 < Ci; k0 += 4) {
    v2f a, bb;
    a.x = pa[0];
    a.y = pa[sA];
    bb.x = pb[0];
    bb.y = pb[1];
    pa += 4 * sA;
    pb += 4;
    acc = mma4(a, bb, acc);
  }
  const int ch = coBase + l16;
  float scale = g[ch] / sqrtf(v[ch] + BN_EPS);
  float shift = b[ch] - m[ch] * scale;
#pragma unroll
  for (int i = 0; i < 8; ++i) {
    int pp = pixBase + i + 8 * half;
    if (pp < HoWo) out[(long)ch * HoWo + pp] = acc[i] * scale + shift;
  }
}

// ---------------------------------------------------------------------------
// side conv: 1x1, 512 -> (channels 223..254 only) + bias; 16x16 spatial.
// ---------------------------------------------------------------------------
__global__ __launch_bounds__(256) void k_side(const float* __restrict__ c2,
                                              const float* __restrict__ w,
                                              const float* __restrict__ bias,
                                              float* __restrict__ feat16) {
  const int oc = blockIdx.x;  // 0..31
  const int p = threadIdx.x;  // 0..255
  const int ch = 223 + oc;
  const float* wp = w + ch * 512;
  float acc = bias[ch];
  for (int ci = 0; ci < 512; ++ci) acc += wp[ci] * c2[ci * 256 + p];
  feat16[oc * 256 + p] = acc;
}

// ---------------------------------------------------------------------------
// bilinear upsample 16x16 -> 512x512 (align-corners linspace semantics)
// ---------------------------------------------------------------------------
__global__ __launch_bounds__(256) void k_upsample(const float* __restrict__ f,
                                                  float* __restrict__ o) {
  long idx = (long)blockIdx.x * blockDim.x + threadIdx.x;
  if (idx >= 32L * 512 * 512) return;
  int j = idx & 511;
  int i = (idx >> 9) & 511;
  int c = (int)(idx >> 18);
  float yv = i * (15.0f / 511.0f);
  float xv = j * (15.0f / 511.0f);
  int y0 = (int)floorf(yv), x0 = (int)floorf(xv);
  int y1 = min(y0 + 1, 15), x1 = min(x0 + 1, 15);
  float wy = yv - y0, wx = xv - x0;
  const float* fp = f + c * 256;
  float t0 = fp[y0 * 16 + x0] * (1.f - wy) + fp[y1 * 16 + x0] * wy;
  float t1 = fp[y0 * 16 + x1] * (1.f - wy) + fp[y1 * 16 + x1] * wy;
  o[idx] = t0 * (1.f - wx) + t1 * wx;
}

__global__ void k_zero(float* p, int n) {
  int i = blockIdx.x * blockDim.x + threadIdx.x;
  if (i < n) p[i] = 0.f;
}

// ---------------------------------------------------------------------------
// sp_feat = sp_maps(101 x P) @ featUp^T(P x 32), WMMA f32 split-K.
// One wave per block; blockIdx.y picks channels 0-15 / 16-31; the wave keeps
// 7 accumulators covering superpixel rows 0..111, reusing the B fragment
// across all 7 M-tiles. Rows >= 101 are CLAMPED (their products land in D
// rows that the epilogue never stores), keeping the inner loop branch-free.
// Split-K over gridDim.x with atomic reduction.
// ---------------------------------------------------------------------------
__global__ __launch_bounds__(32) void k_spfeat_wmma(
    const float* __restrict__ sp, const float* __restrict__ feat,
    float* __restrict__ spf, int P, int N) {
  const int lane = threadIdx.x & 31;
  const int half = lane >> 4;
  const int l16 = lane & 15;
  const int cbase = blockIdx.y * 16;
  const int chunk = P / gridDim.x;
  const int p0 = blockIdx.x * chunk;

  v8f acc[7];
#pragma unroll
  for (int t = 0; t < 7; ++t)
    acc[t] = (v8f){0.f, 0.f, 0.f, 0.f, 0.f, 0.f, 0.f, 0.f};

  const float* rows[7];
#pragma unroll
  for (int t = 0; t < 7; ++t) {
    int n = t * 16 + l16;
    rows[t] = sp + (long)min(n, N - 1) * P;  // clamped: junk rows discarded
  }
  const float* fp = feat + (long)(cbase + l16) * P + p0 + half * 2;
  const float* ap0 = p0 + (const float*)0;  // (kept simple below)

  for (int k = 0; k < chunk; k += 4) {
    int pb = p0 + k + half * 2;
    v2f bb;
    bb.x = fp[0];
    bb.y = fp[1];
    fp += 4;
#pragma unroll
    for (int t = 0; t < 7; ++t) {
      v2f a;
      a.x = rows[t][pb];
      a.y = rows[t][pb + 1];
      acc[t] = mma4(a, bb, acc[t]);
    }
  }
  (void)ap0;
#pragma unroll
  for (int t = 0; t < 7; ++t)
#pragma unroll
    for (int i = 0; i < 8; ++i) {
      int n = t * 16 + i + 8 * half;
      if (n < N) atomicAdd(&spf[n * 32 + cbase + l16], acc[t][i]);
    }
}

// ---------------------------------------------------------------------------
// logits + softmax (101 x 2) -> probs of class 1
// ---------------------------------------------------------------------------
__global__ void k_logits(const float* __restrict__ spf,
                         const float* __restrict__ cw,
                         const float* __restrict__ cb,
                         float* __restrict__ probs1, int N) {
  int n = threadIdx.x;
  if (n >= N) return;
  float l0 = cb[0], l1 = cb[1];
  for (int c = 0; c < 32; ++c) {
    float f = spf[n * 32 + c];
    l0 += f * cw[c];
    l1 += f * cw[32 + c];
  }
  float mx = fmaxf(l0, l1);
  float e0 = __expf(l0 - mx), e1 = __expf(l1 - mx);
  probs1[n] = e1 / (e0 + e1);
}

// per-pixel argmax over superpixel maps + gather of class-1 probability
__global__ __launch_bounds__(256) void k_predict(
    const float* __restrict__ sp, const float* __restrict__ probs1,
    float* __restrict__ out, int P, int N) {
  int p = blockIdx.x * blockDim.x + threadIdx.x;
  if (p >= P) return;
  float best = sp[p];
  int bi = 0;
  for (int n = 1; n < N; ++n) {
    float vv = sp[(long)n * P + p];
    if (vv > best) { best = vv; bi = n; }  // first max wins (jnp.argmax)
  }
  out[p] = probs1[bi];
}

// ===========================================================================
extern "C" void kernel_launch(void* const* d_in, const int* in_sizes, int n_in,
                              void* d_out, int out_size, void* d_ws,
                              size_t ws_size, hipStream_t stream) {
  (void)in_sizes; (void)n_in; (void)out_size; (void)ws_size;
  int ii = 0;
  auto nf = [&]() { return (const float*)d_in[ii++]; };

  struct BN { const float *g, *b, *m, *v; };
  auto nbn = [&]() {
    BN r; r.g = nf(); r.b = nf(); r.m = nf(); r.v = nf(); return r;
  };

  const float* x = nf();   // (1,3,512,512)
  const float* sp = nf();  // (101,512,512)
  const float* w7 = nf();  // conv1 (64,3,7,7)
  BN bn1 = nbn();

  struct Blk { const float* w1; BN b1; const float* w2; BN b2;
               const float* dw; BN dbn; bool down; };
  Blk blk[4][2];
  for (int si = 0; si < 4; ++si)
    for (int bi = 0; bi < 2; ++bi) {
      Blk& B = blk[si][bi];
      B.w1 = nf(); B.b1 = nbn(); B.w2 = nf(); B.b2 = nbn();
      B.down = (si > 0 && bi == 0);
      if (B.down) { B.dw = nf(); B.dbn = nbn(); }
      else { B.dw = nullptr; }
    }
  const float* side_w = nf();  // (256,512)
  const float* side_b = nf();  // (256,)
  const float* cls_w = nf();   // (2,32)
  const float* cls_b = nf();   // (2,)

  float* ws = (float*)d_ws;
  float* A0  = ws;             // 64*256*256
  float* A1  = ws + 4194304;   // 64*128*128
  float* P0  = ws + 5242880;   // 1M
  float* P1  = ws + 6291456;   // 1M
  float* PD  = ws + 7340032;   // downsample identity
  float* C2  = ws + 8388608;   // 512*16*16 (last conv2, pre-BN)
  float* F16 = ws + 8519680;   // 32*16*16
  float* FUP = ws + 8527872;   // 32*512*512
  float* SPF = ws + 16916480;  // 101*32
  float* PRB = ws + 16919712;  // 101
  float* ZB  = ws + 16919840;  // 16-float zero page for OOB lanes

  // zero page + split-K accumulator init
  k_zero<<<1, 64, 0, stream>>>(ZB, 16);
  k_zero<<<13, 256, 0, stream>>>(SPF, 101 * 32);

  // stem
  k_conv7_bn_relu<<<dim3(16, 16, 64), dim3(16, 16), 0, stream>>>(
      x, w7, bn1.g, bn1.b, bn1.m, bn1.v, A0);
  k_maxpool<<<4096, 256, 0, stream>>>(A0, A1);

  // residual stages
  const int stage_c[4] = {64, 128, 256, 512};
  float* pool[3] = {A1, P0, P1};
  float* cur = A1;
  int curC = 64, curS = 128;
  for (int si = 0; si < 4; ++si) {
    int Co = stage_c[si];
    for (int bi = 0; bi < 2; ++bi) {
      Blk& B = blk[si][bi];
      int stride = B.down ? 2 : 1;
      int H = curS, Ho = curS / stride;
      int HoWo = Ho * Ho;
      float *f1 = nullptr, *f2 = nullptr;
      for (int q = 0; q < 3; ++q)
        if (pool[q] != cur) { if (!f1) f1 = pool[q]; else f2 = pool[q]; }
      dim3 grid((HoWo + 127) / 128, Co / 16);

      // out = relu(bn1(conv1(h)))
      k_conv3_wmma<<<grid, 256, 0, stream>>>(
          cur, B.w1, B.b1.g, B.b1.b, B.b1.m, B.b1.v, nullptr, f1, nullptr, ZB,
          curC, H, H, Co, Ho, Ho, stride, 1);
      // identity path
      const float* idn = cur;
      if (B.down) {
        k_conv1_wmma<<<grid, 256, 0, stream>>>(
            cur, B.dw, B.dbn.g, B.dbn.b, B.dbn.m, B.dbn.v, PD, ZB, curC, H, H,
            Co, Ho, Ho, stride);
        idn = PD;
      }
      // h = relu(bn2(conv2(out)) + idn); save raw conv2 of the final block
      float* raw = (si == 3 && bi == 1) ? C2 : nullptr;
      k_conv3_wmma<<<grid, 256, 0, stream>>>(
          f1, B.w2, B.b2.g, B.b2.b, B.b2.m, B.b2.v, idn, f2, raw, ZB, Co, Ho,
          Ho, Co, Ho, Ho, 1, 1);
      cur = f2;
      curC = Co;
      curS = Ho;
    }
  }

  // head
  k_side<<<32, 256, 0, stream>>>(C2, side_w, side_b, F16);
  k_upsample<<<32768, 256, 0, stream>>>(F16, FUP);
  k_spfeat_wmma<<<dim3(256, 2), 32, 0, stream>>>(sp, FUP, SPF, 512 * 512, 101);
  k_logits<<<1, 128, 0, stream>>>(SPF, cls_w, cls_b, PRB, 101);
  k_predict<<<1024, 256, 0, stream>>>(sp, PRB, (float*)d_out, 512 * 512, 101);
}